// KGCN_26628797235223
// MI455X (gfx1250) — compile-verified
//
#include <hip/hip_runtime.h>

typedef float v2f __attribute__((ext_vector_type(2)));
typedef float v8f __attribute__((ext_vector_type(8)));

__device__ __forceinline__ v8f wmma_f32(v2f a, v2f b, v8f c) {
  // V_WMMA_F32_16X16X4_F32: A 16x4 f32 (2 VGPRs), B 4x16 f32 (2 VGPRs), C/D 16x16 f32 (8 VGPRs)
  return __builtin_amdgcn_wmma_f32_16x16x4_f32(false, a, false, b, (short)0, c, false, false);
}

__device__ __forceinline__ void wbar() { __builtin_amdgcn_wave_barrier(); }

// C/D layout: VGPR r, lanes 0-15 -> (M=r, N=lane); lanes 16-31 -> (M=r+8, N=lane-16)
__device__ __forceinline__ void store_c_lds(v8f c, float* T, int lane) {
  int n = lane & 15, hi = lane >> 4;
#pragma unroll
  for (int r = 0; r < 8; ++r) T[(r + 8 * hi) * 17 + n] = c[r];
}

// One dense layer 16 -> DOUT from an LDS-resident 16x16 activation tile (stride 17).
// Branchless: clamp the output-column index and zero-select, so no EXEC-masked loads.
template <int DOUT, bool RELU>
__device__ __forceinline__ v8f dense16_lds(const float* T, const float* __restrict__ W,
                                           const float* __restrict__ bias, int lane) {
  int m = lane & 15, hi = lane >> 4, n = m;
  int nc = (DOUT < 16) ? ((n < DOUT) ? n : (DOUT - 1)) : n;
  bool nvalid = (DOUT < 16) ? (n < DOUT) : true;
  v8f c = {};
#pragma unroll
  for (int s = 0; s < 4; ++s) {
    int k = 4 * s + 2 * hi;  // A/B element (VGPR j, lane L): K = 4s + 2*(L>>4) + j
    v2f a, b;
    a.x = T[m * 17 + k];
    a.y = T[m * 17 + k + 1];
    float b0 = W[(k + 0) * DOUT + nc];
    float b1 = W[(k + 1) * DOUT + nc];
    b.x = nvalid ? b0 : 0.f;
    b.y = nvalid ? b1 : 0.f;
    c = wmma_f32(a, b, c);
  }
  float bv = nvalid ? bias[nc] : 0.f;
#pragma unroll
  for (int r = 0; r < 8; ++r) {
    c[r] += bv;
    if (RELU) c[r] = fmaxf(c[r], 0.f);
  }
  return c;
}

// First edge layer: A row = concat(x[src] (DX), edge_attr (DE), x[dst] (DX)), K padded to mult of 4.
// Region select is branchless (cndmask on the address); only the compile-time tail step pads.
template <int DX, int DE>
__device__ __forceinline__ v8f edge_layer1(const float* __restrict__ x, const float* __restrict__ ea,
                                           int sidx, int didx, long erow,
                                           const float* __restrict__ W, const float* __restrict__ bias,
                                           int lane) {
  constexpr int K1 = 2 * DX + DE;
  constexpr int KP = (K1 + 3) & ~3;
  int hi = lane >> 4, n = lane & 15;
  const float* xs = x + (long)sidx * DX;
  const float* xd = x + (long)didx * DX;
  const float* ee = ea + erow * DE;
  v8f c = {};
#pragma unroll
  for (int s = 0; s < KP / 4; ++s) {
    v2f a, b;
#pragma unroll
    for (int j = 0; j < 2; ++j) {
      int f = 4 * s + 2 * hi + j;
      int fd = f - DX - DE;
      int fdc = (fd < DX) ? fd : (DX - 1);  // clamp pad lanes to a safe address
      const float* p = (f < DX) ? (xs + f) : (f < DX + DE) ? (ee + (f - DX)) : (xd + fdc);
      float av = p[0];
      int wk = (f < K1) ? f : (K1 - 1);
      float bv = W[wk * 16 + n];
      if (4 * s + 3 >= K1) {  // compile-time: only the tail step can pad
        bool valid = (f < K1);
        av = valid ? av : 0.f;
        bv = valid ? bv : 0.f;
      }
      a[j] = av;
      b[j] = bv;
    }
    c = wmma_f32(a, b, c);
  }
  float bb = bias[n];
#pragma unroll
  for (int r = 0; r < 8; ++r) c[r] = fmaxf(c[r] + bb, 0.f);
  return c;
}

// Edge conv: e_out = MLP(concat(x[src], ea, x[dst])); atomically aggregate concat(x[src], e_out) at dst.
template <int DX, int DE, int DOUT, bool THREE, bool FRELU>
__global__ __launch_bounds__(128) void edge_conv_kernel(
    const float* __restrict__ x, const float* __restrict__ ea,
    const int* __restrict__ src, const int* __restrict__ dst,
    const float* __restrict__ W1, const float* __restrict__ b1,
    const float* __restrict__ W2, const float* __restrict__ b2,
    const float* __restrict__ W3, const float* __restrict__ b3,
    float* __restrict__ eout, float* __restrict__ agg, int n_edges) {
  __shared__ float tiles[4][16 * 17];
  int wave = threadIdx.x >> 5, lane = threadIdx.x & 31;
  long base = ((long)blockIdx.x * 4 + wave) * 16;
  if (base >= n_edges) return;  // wave-uniform; counts are multiples of 16 so no partial tiles
  float* T = tiles[wave];
  int m = lane & 15;
  long erow = base + m;
  int sidx = src[erow], didx = dst[erow];

  v8f h = edge_layer1<DX, DE>(x, ea, sidx, didx, erow, W1, b1, lane);
  store_c_lds(h, T, lane);
  wbar();
  v8f e;
  if constexpr (THREE) {
    v8f h2 = dense16_lds<16, true>(T, W2, b2, lane);
    wbar();
    store_c_lds(h2, T, lane);
    wbar();
    e = dense16_lds<DOUT, FRELU>(T, W3, b3, lane);
  } else {
    e = dense16_lds<DOUT, FRELU>(T, W2, b2, lane);
  }
  wbar();
  store_c_lds(e, T, lane);
  wbar();

  constexpr int AW = DX + DOUT;

  // e-part: coalesced store of the 16 x DOUT edge-repr tile, fused with its scatter-add.
#pragma unroll 1
  for (int idx = lane; idx < 16 * DOUT; idx += 32) {
    int mm = idx / DOUT, f = idx - mm * DOUT;
    float v = T[mm * 17 + f];
    eout[base * DOUT + idx] = v;
    atomicAdd(&agg[(long)dst[base + mm] * AW + DX + f], v);
  }

  // x-part of the message: gather x[src] and scatter-add (uniform loop, no divergence).
#pragma unroll 1
  for (int idx = lane; idx < 16 * DX; idx += 32) {
    int mm = idx / DX, f = idx - mm * DX;
    long e2 = base + mm;
    atomicAdd(&agg[(long)dst[e2] * AW + f], x[(long)src[e2] * DX + f]);
  }
}

// Node conv: x_out = MLP(agg), agg rows are AW wide.
template <int AW, int DOUT, bool THREE, bool FRELU>
__global__ __launch_bounds__(128) void node_conv_kernel(
    const float* __restrict__ agg,
    const float* __restrict__ W1, const float* __restrict__ b1,
    const float* __restrict__ W2, const float* __restrict__ b2,
    const float* __restrict__ W3, const float* __restrict__ b3,
    float* __restrict__ xout, int n_nodes) {
  __shared__ float tiles[4][16 * 17];
  int wave = threadIdx.x >> 5, lane = threadIdx.x & 31;
  long base = ((long)blockIdx.x * 4 + wave) * 16;
  if (base >= n_nodes) return;
  float* T = tiles[wave];
  int m = lane & 15, hi = lane >> 4, n = m;

  constexpr int KP = (AW + 3) & ~3;
  const float* arow = agg + (base + m) * AW;
  v8f c = {};
#pragma unroll
  for (int s = 0; s < KP / 4; ++s) {
    v2f a, b;
#pragma unroll
    for (int j = 0; j < 2; ++j) {
      int f = 4 * s + 2 * hi + j;
      int fc = (f < AW) ? f : (AW - 1);  // clamp pad lanes, zero-select below
      float av = arow[fc];
      float bv = W1[fc * 16 + n];
      if (4 * s + 3 >= AW) {  // compile-time: only the tail step can pad
        bool valid = (f < AW);
        av = valid ? av : 0.f;
        bv = valid ? bv : 0.f;
      }
      a[j] = av;
      b[j] = bv;
    }
    c = wmma_f32(a, b, c);
  }
  float bb = b1[n];
#pragma unroll
  for (int r = 0; r < 8; ++r) c[r] = fmaxf(c[r] + bb, 0.f);

  store_c_lds(c, T, lane);
  wbar();
  v8f o;
  if constexpr (THREE) {
    v8f h2 = dense16_lds<16, true>(T, W2, b2, lane);
    wbar();
    store_c_lds(h2, T, lane);
    wbar();
    o = dense16_lds<DOUT, FRELU>(T, W3, b3, lane);
  } else {
    o = dense16_lds<DOUT, FRELU>(T, W2, b2, lane);
  }
  wbar();
  store_c_lds(o, T, lane);
  wbar();
#pragma unroll 1
  for (int idx = lane; idx < 16 * DOUT; idx += 32)
    xout[base * DOUT + idx] = T[(idx / DOUT) * 17 + (idx % DOUT)];
}

__global__ void zero_kernel(float* __restrict__ p, long n) {
  long i = (long)blockIdx.x * blockDim.x + threadIdx.x;
  long stride = (long)gridDim.x * blockDim.x;
  for (; i < n; i += stride) p[i] = 0.f;
}

extern "C" void kernel_launch(void* const* d_in, const int* in_sizes, int n_in,
                              void* d_out, int out_size, void* d_ws, size_t ws_size,
                              hipStream_t stream) {
  const float* x_node = (const float*)d_in[0];
  const float* x_edge = (const float*)d_in[1];
  const int* eidx = (const int*)d_in[2];
  int n_nodes = in_sizes[0] / 12;
  int n_edges = in_sizes[2] / 2;
  const int* src = eidx;
  const int* dst = eidx + n_edges;

  // params flattened alphabetically: c1_edge, c1_node, c2_edge, c2_node, c3_edge, c3_node;
  // each layer contributes (W, b) leaves in order.
  const float* P[28];
  for (int i = 0; i < 28; ++i) P[i] = (const float*)d_in[3 + i];

  float* xbuf = (float*)d_ws;                  // n_nodes * 16
  float* agg = xbuf + (size_t)n_nodes * 16;    // n_nodes * 32 (max width)
  float* ebuf = agg + (size_t)n_nodes * 32;    // n_edges * 16
  float* out_x = (float*)d_out;                // n_nodes * 3
  float* out_e = out_x + (size_t)n_nodes * 3;  // n_edges * 3

  int tilesE = (n_edges + 15) / 16, blocksE = (tilesE + 3) / 4;
  int tilesN = (n_nodes + 15) / 16, blocksN = (tilesN + 3) / 4;
  dim3 blk(128);

  // ---- conv1: edge MLP [30,16,16], node MLP [28,16,16], relu ----
  zero_kernel<<<512, 256, 0, stream>>>(agg, (long)n_nodes * 28);
  edge_conv_kernel<12, 6, 16, false, true><<<blocksE, blk, 0, stream>>>(
      x_node, x_edge, src, dst, P[0], P[1], P[2], P[3], nullptr, nullptr, ebuf, agg, n_edges);
  node_conv_kernel<28, 16, false, true><<<blocksN, blk, 0, stream>>>(
      agg, P[4], P[5], P[6], P[7], nullptr, nullptr, xbuf, n_nodes);

  // ---- conv2: edge MLP [48,16,16], node MLP [32,16,16], relu ----
  zero_kernel<<<512, 256, 0, stream>>>(agg, (long)n_nodes * 32);
  edge_conv_kernel<16, 16, 16, false, true><<<blocksE, blk, 0, stream>>>(
      xbuf, ebuf, src, dst, P[8], P[9], P[10], P[11], nullptr, nullptr, ebuf, agg, n_edges);
  node_conv_kernel<32, 16, false, true><<<blocksN, blk, 0, stream>>>(
      agg, P[12], P[13], P[14], P[15], nullptr, nullptr, xbuf, n_nodes);

  // ---- conv3: edge MLP [48,16,16,3], node MLP [19,16,16,3], no final relu ----
  zero_kernel<<<512, 256, 0, stream>>>(agg, (long)n_nodes * 19);
  edge_conv_kernel<16, 16, 3, true, false><<<blocksE, blk, 0, stream>>>(
      xbuf, ebuf, src, dst, P[16], P[17], P[18], P[19], P[20], P[21], out_e, agg, n_edges);
  node_conv_kernel<19, 3, true, false><<<blocksN, blk, 0, stream>>>(
      agg, P[22], P[23], P[24], P[25], P[26], P[27], out_x, n_nodes);
}